// SpectralConv2d_64587718197750
// MI455X (gfx1250) — compile-verified
//
#include <hip/hip_runtime.h>

typedef float v2f __attribute__((ext_vector_type(2)));
typedef float v8f __attribute__((ext_vector_type(8)));

#define TWO_PI 6.28318530717958647692f

// workspace float offsets
#define OFF_F2T   0u                       // [32][256]   fwd ky twiddles (B-layout, transposed)
#define OFF_WBIG  8192u                    // [64][512]   fwd kx twiddles (A-layout)
#define OFF_WINV  40960u                   // [256][64]   inv kx twiddles (A-layout)
#define OFF_GT    57344u                   // [256][32]   inv ky twiddles (B-layout, transposed)
#define OFF_T     65536u                   // [512*256][32]  stage-A output (Tr cols 0-15 | Ti cols 16-31)
#define OFF_Y     (OFF_T + 4194304u)       // [512][64][16]  modes in  (Yr rows 0-31, Yi rows 32-63)
#define OFF_O     (OFF_Y + 524288u)        // [512][64][16]  modes out (Or rows 0-31, Oi rows 32-63)
#define OFF_Z2    (OFF_O + 524288u)        // [512][256][32] stage-D output (Zr cols 0-15, Zi cols 16-31)

static __device__ __forceinline__ v8f wmma4(v2f a, v2f b, v8f c) {
  // D = A(16x4 f32) * B(4x16 f32) + C(16x16 f32)
  return __builtin_amdgcn_wmma_f32_16x16x4_f32(false, a, false, b, (short)0, c, false, false);
}

static __device__ __forceinline__ float ang(int num) {
  return TWO_PI * (float)(num & 255) * (1.0f / 256.0f);
}

// ---------------------------------------------------------------- tables
__global__ void init_tables(float* __restrict__ ws) {
  int tid = blockIdx.x * blockDim.x + threadIdx.x;
  int stride = gridDim.x * blockDim.x;
  // F2t [n=32][k=w=256]: n<16 -> cos(2*pi*n*w/256), n>=16 -> -sin(...)  (rfft along w)
  for (int i = tid; i < 32 * 256; i += stride) {
    int n = i >> 8, w = i & 255;
    float th = ang((n & 15) * w);
    ws[OFF_F2T + i] = (n < 16) ? cosf(th) : -sinf(th);
  }
  // Wbig [m=64][k=512]: rows 0-31 produce Yr, 32-63 produce Yi; cols 0-255 hit Tr, 256-511 hit Ti
  for (int i = tid; i < 64 * 512; i += stride) {
    int m = i >> 9, k = i & 511;
    int h = k & 255, part = k >> 8;
    int mm = m & 31;
    int fm = (mm < 16) ? mm : (224 + mm);   // kx = 0..15 then 240..255
    float th = ang(fm * h);
    float c = cosf(th), s = sinf(th);
    ws[OFF_WBIG + i] = (m < 32) ? (part ? s : c) : (part ? c : -s);
  }
  // WinvA [h=256][k=64]: k<32 -> cos(+2*pi*f(k)*h/256), k>=32 -> sin(...)
  for (int i = tid; i < 256 * 64; i += stride) {
    int h = i >> 6, k = i & 63;
    int m = k & 31;
    int fm = (m < 16) ? m : (224 + m);
    float th = ang(fm * h);
    ws[OFF_WINV + i] = (k < 32) ? cosf(th) : sinf(th);
  }
  // Gt [w=256][k=32]: y += Zr*a*cos - Zi*a*sin, a = (ky?2:1)/65536
  for (int i = tid; i < 256 * 32; i += stride) {
    int w = i >> 5, k = i & 31;
    int ky = k & 15;
    float th = ang(ky * w);
    float a = ((ky == 0) ? 1.0f : 2.0f) * (1.0f / 65536.0f);
    ws[OFF_GT + i] = (k < 16) ? a * cosf(th) : -a * sinf(th);
  }
}

// ------------- stage A: T[131072x32] = x[131072x256] @ F2t^T ; one wave = 16x32 tile, A shared by 2 WMMAs
__global__ void __launch_bounds__(128) stageA(const float* __restrict__ x, float* __restrict__ ws) {
  const float* F2t = ws + OFF_F2T;
  float* T = ws + OFF_T;
  int wid  = blockIdx.x * (blockDim.x >> 5) + (threadIdx.x >> 5);   // rowTile, 8192 total
  int lane = threadIdx.x & 31, lm = lane & 15, half = lane >> 4;
  int r = (wid << 4) + lm;
  const float* arow  = x + (size_t)r * 256 + 2 * half;
  const float* brow0 = F2t + lm * 256 + 2 * half;          // n = 0..15  (cos)
  const float* brow1 = brow0 + 16 * 256;                   // n = 16..31 (-sin)
  v8f acc0 = {}, acc1 = {};
#pragma unroll 4
  for (int k0 = 0; k0 < 256; k0 += 4) {
    if ((k0 & 127) == 0) __builtin_prefetch(arow + k0 + 128, 0, 1);
    v2f a  = *(const v2f*)(arow + k0);
    v2f b0 = *(const v2f*)(brow0 + k0);
    v2f b1 = *(const v2f*)(brow1 + k0);
    acc0 = wmma4(a, b0, acc0);
    acc1 = wmma4(a, b1, acc1);
  }
  int ob = ((wid << 4) + 8 * half) * 32 + lm;
#pragma unroll
  for (int v = 0; v < 8; ++v) {
    T[ob + v * 32]      = acc0[v];
    T[ob + v * 32 + 16] = acc1[v];
  }
}

// ------------- stage B: Y_p[64x16] = Wbig[64x512] @ [Tr;Ti]_p ; one wave per pair, B shared by 4 WMMAs
__global__ void __launch_bounds__(128) stageB(float* __restrict__ ws) {
  const float* Wb = ws + OFF_WBIG;
  const float* T  = ws + OFF_T;
  float* Y = ws + OFF_Y;
  int p    = blockIdx.x * (blockDim.x >> 5) + (threadIdx.x >> 5);   // pair, 512 total
  int lane = threadIdx.x & 31, lm = lane & 15, half = lane >> 4;
  const float* a0 = Wb + lm * 512 + 2 * half;              // m-tiles 0..3 share column k
  const float* Tp = T + p * 8192;
  v8f acc0 = {}, acc1 = {}, acc2 = {}, acc3 = {};
#pragma unroll 2
  for (int k0 = 0; k0 < 512; k0 += 4) {
    int k = k0 + 2 * half;
    int idx = (k < 256) ? (k * 32 + lm) : ((k - 256) * 32 + 16 + lm);
    v2f b;
    b.x = Tp[idx];
    b.y = Tp[idx + 32];
    acc0 = wmma4(*(const v2f*)(a0 + k0),             b, acc0);
    acc1 = wmma4(*(const v2f*)(a0 + 16 * 512 + k0),  b, acc1);
    acc2 = wmma4(*(const v2f*)(a0 + 32 * 512 + k0),  b, acc2);
    acc3 = wmma4(*(const v2f*)(a0 + 48 * 512 + k0),  b, acc3);
  }
  int yb = p * 1024 + 8 * half * 16 + lm;
#pragma unroll
  for (int v = 0; v < 8; ++v) {
    Y[yb + v * 16]            = acc0[v];
    Y[yb + v * 16 + 16 * 16]  = acc1[v];
    Y[yb + v * 16 + 32 * 16]  = acc2[v];
    Y[yb + v * 16 + 48 * 16]  = acc3[v];
  }
}

// ------------- mix: per-mode 32x32 complex channel mix (VALU; ~1% of FLOPs)
__global__ void __launch_bounds__(256) mixmodes(const float* __restrict__ w1,
                                               const float* __restrict__ w2,
                                               float* __restrict__ ws) {
  const float* Y = ws + OFF_Y;
  float* O = ws + OFF_O;
  int tid = blockIdx.x * blockDim.x + threadIdx.x;
  int ky = tid & 15, o = (tid >> 4) & 31, kx = (tid >> 9) & 31, b = tid >> 14;
  const float* wsel = (kx < 16) ? w1 : w2;
  int kxi = kx & 15;
  float Or = 0.0f, Oi = 0.0f;
#pragma unroll 4
  for (int i = 0; i < 32; ++i) {
    int pb = (b * 32 + i) * 1024;
    float yr = Y[pb + kx * 16 + ky];
    float yi = Y[pb + (32 + kx) * 16 + ky];
    int wix = (((i * 32 + o) * 16 + kxi) * 16 + ky) * 2;
    float wr = wsel[wix], wi = wsel[wix + 1];
    Or += yr * wr - yi * wi;
    Oi += yr * wi + yi * wr;
  }
  int qb = (b * 32 + o) * 1024;
  O[qb + kx * 16 + ky] = Or;
  O[qb + (32 + kx) * 16 + ky] = Oi;
}

// ------------- stage D: Z_q[256x32] = WinvA[256x64] @ Bd(O_q) ; one wave = 16x32 (Zr+Zi share O loads)
__global__ void __launch_bounds__(128) stageD(float* __restrict__ ws) {
  const float* Wi = ws + OFF_WINV;
  const float* O  = ws + OFF_O;
  float* Z = ws + OFF_Z2;
  int wid  = blockIdx.x * (blockDim.x >> 5) + (threadIdx.x >> 5);   // 8192 total
  int lane = threadIdx.x & 31, lm = lane & 15, half = lane >> 4;
  int q = wid >> 4, mt = wid & 15;
  const float* arow = Wi + ((mt << 4) + lm) * 64 + 2 * half;
  const float* Oq = O + q * 1024;
  v8f acc0 = {}, acc1 = {};
#pragma unroll
  for (int k0 = 0; k0 < 64; k0 += 4) {
    v2f a = *(const v2f*)(arow + k0);
    int kb = k0 + 2 * half;
    v2f b0, b1;   // b0: Zr column (cos*Or - sin*Oi) ; b1: Zi column (cos*Oi + sin*Or)
#pragma unroll
    for (int v = 0; v < 2; ++v) {
      int k = kb + v;
      int m = k & 31;
      float o_r = Oq[m * 16 + lm];
      float o_i = Oq[(32 + m) * 16 + lm];
      b0[v] = (k < 32) ? o_r : -o_i;
      b1[v] = (k < 32) ? o_i :  o_r;
    }
    acc0 = wmma4(a, b0, acc0);
    acc1 = wmma4(a, b1, acc1);
  }
  int zb = q * 8192 + ((mt << 4) + 8 * half) * 32 + lm;
#pragma unroll
  for (int v = 0; v < 8; ++v) {
    Z[zb + v * 32]      = acc0[v];
    Z[zb + v * 32 + 16] = acc1[v];
  }
}

// ------------- stage E: y_q[256x256] = Z_q[256x32] @ Gt^T ; one wave = 16x64 (4 w-tiles share A)
__global__ void __launch_bounds__(128) stageE(float* __restrict__ out, const float* __restrict__ ws) {
  const float* Gt = ws + OFF_GT;
  const float* Z  = ws + OFF_Z2;
  int wid  = blockIdx.x * (blockDim.x >> 5) + (threadIdx.x >> 5);   // 32768 total
  int lane = threadIdx.x & 31, lm = lane & 15, half = lane >> 4;
  int q = wid >> 6, rest = wid & 63;
  int h0 = (rest >> 2) << 4;          // h-tile 0..15
  int w0 = (rest & 3) << 6;           // w-group of 4 tiles: 0,64,128,192
  const float* arow = Z + q * 8192 + (h0 + lm) * 32 + 2 * half;
  const float* b0r = Gt + (w0 + lm) * 32 + 2 * half;
  v8f acc0 = {}, acc1 = {}, acc2 = {}, acc3 = {};
#pragma unroll
  for (int k0 = 0; k0 < 32; k0 += 4) {
    v2f a = *(const v2f*)(arow + k0);
    acc0 = wmma4(a, *(const v2f*)(b0r + k0),            acc0);
    acc1 = wmma4(a, *(const v2f*)(b0r + 16 * 32 + k0),  acc1);
    acc2 = wmma4(a, *(const v2f*)(b0r + 32 * 32 + k0),  acc2);
    acc3 = wmma4(a, *(const v2f*)(b0r + 48 * 32 + k0),  acc3);
  }
  size_t ob = (size_t)q * 65536 + (size_t)(h0 + 8 * half) * 256 + w0 + lm;
#pragma unroll
  for (int v = 0; v < 8; ++v) {
    out[ob + (size_t)v * 256]      = acc0[v];
    out[ob + (size_t)v * 256 + 16] = acc1[v];
    out[ob + (size_t)v * 256 + 32] = acc2[v];
    out[ob + (size_t)v * 256 + 48] = acc3[v];
  }
}

extern "C" void kernel_launch(void* const* d_in, const int* in_sizes, int n_in,
                              void* d_out, int out_size, void* d_ws, size_t ws_size,
                              hipStream_t stream) {
  (void)in_sizes; (void)n_in; (void)out_size; (void)ws_size;
  const float* x  = (const float*)d_in[0];
  const float* w1 = (const float*)d_in[1];
  const float* w2 = (const float*)d_in[2];
  float* out = (float*)d_out;
  float* ws  = (float*)d_ws;

  init_tables<<<64, 256, 0, stream>>>(ws);
  stageA<<<2048, 128, 0, stream>>>(x, ws);      // 8192 waves,  128 wmma each (x read once)
  stageB<<<128, 128, 0, stream>>>(ws);          // 512 waves,   512 wmma each (T read once)
  mixmodes<<<1024, 256, 0, stream>>>(w1, w2, ws);
  stageD<<<2048, 128, 0, stream>>>(ws);         // 8192 waves,   32 wmma each (O read once)
  stageE<<<8192, 128, 0, stream>>>(out, ws);    // 32768 waves,  32 wmma each (Z read 1x per 4 tiles)
}